// Dilation2d_4973572128825
// MI455X (gfx1250) — compile-verified
//
#include <hip/hip_runtime.h>
#include <cstdint>

// Problem shape (fixed by the reference)
#define N_   8
#define CI_  4
#define CO_  8
#define H_   512
#define W_   512
#define KH_  5
#define KW_  5
#define PAD_ 2

// Tile shape
#define TW_  64
#define TH_  16
#define PW_  (TW_ + KW_ - 1)   // 68
#define PH_  (TH_ + KH_ - 1)   // 20
#define NTHREADS 256

__global__ __launch_bounds__(NTHREADS)
void dilation2d_maxplus_kernel(const float* __restrict__ x,
                               const float* __restrict__ wt,
                               float* __restrict__ out)
{
    // x tile: 4 ch x 20 x 68 floats = 21.25 KB ; weights: 800 floats = 3.125 KB
    // row stride 272B and channel stride 5440B are both 16B-aligned -> float4 LDS reads OK
    __shared__ __align__(16) float sx[CI_][PH_][PW_];
    __shared__ __align__(16) float sw[CI_][KH_][KW_][CO_];

    const int tid = threadIdx.x;
    const int n   = blockIdx.z;
    const int h0  = blockIdx.y * TH_;
    const int w0  = blockIdx.x * TW_;

    const float NEG_INF = -__builtin_huge_valf();

    // ---- weights: wt[co][ci][kh][kw] -> sw[ci][kh][kw][co] (co contiguous for b128 broadcast)
    for (int idx = tid; idx < CO_ * CI_ * KH_ * KW_; idx += NTHREADS) {
        int co = idx / (CI_ * KH_ * KW_);
        int r  = idx % (CI_ * KH_ * KW_);
        ((float*)sw)[r * CO_ + co] = wt[idx];
    }

    // ---- pre-fill x tile with -inf (implements the -inf pad for lanes that skip the load)
    for (int idx = tid; idx < CI_ * PH_ * PW_; idx += NTHREADS) {
        ((float*)sx)[idx] = NEG_INF;
    }
    __syncthreads();

    // ---- CDNA5 async global->LDS copy of the in-bounds window (EXEC-masked at borders)
    for (int idx = tid; idx < CI_ * PH_ * PW_; idx += NTHREADS) {
        int ci  = idx / (PH_ * PW_);
        int r   = idx % (PH_ * PW_);
        int row = r / PW_;
        int col = r % PW_;
        int gh  = h0 + row - PAD_;
        int gw  = w0 + col - PAD_;
        if ((unsigned)gh < (unsigned)H_ && (unsigned)gw < (unsigned)W_) {
            // low 32 bits of a generic pointer into LDS == wave-relative LDS byte address
            uint32_t lds = (uint32_t)(uintptr_t)(&sx[ci][row][col]);
            uint64_t ga  = (uint64_t)(uintptr_t)(x + (((size_t)n * CI_ + ci) * H_ + gh) * W_ + gw);
            asm volatile("global_load_async_to_lds_b32 %0, %1, off"
                         :: "v"(lds), "v"(ga)
                         : "memory");
        }
    }
    // wait for this wave's async LDS writes, then make them visible workgroup-wide
    asm volatile("s_wait_asynccnt 0" ::: "memory");
    __syncthreads();

    // ---- compute: thread owns 4 consecutive cols (txq), 1 row (ty), all 8 co
    const int txq = tid & 15;        // column quad: cols 4*txq .. 4*txq+3
    const int ty  = tid >> 4;        // row 0..15

    float acc[CO_][4];
    #pragma unroll
    for (int c = 0; c < CO_; ++c)
        #pragma unroll
        for (int j = 0; j < 4; ++j)
            acc[c][j] = NEG_INF;

    #pragma unroll 1
    for (int ci = 0; ci < CI_; ++ci) {
        #pragma unroll 1
        for (int kh = 0; kh < KH_; ++kh) {
            // 5 kw x 8 co weights for this (ci,kh): 40 contiguous floats -> 10 float4 broadcasts
            const float4* wp = (const float4*)(&sw[ci][kh][0][0]);
            float4 wq[10];
            #pragma unroll
            for (int i = 0; i < 10; ++i) wq[i] = wp[i];

            // 8-float sliding window for 4 output pixels: two aligned b128 LDS reads
            const float4* xp = (const float4*)(&sx[ci][ty + kh][4 * txq]);
            float4 xa = xp[0];
            float4 xb = xp[1];
            float xr[8] = {xa.x, xa.y, xa.z, xa.w, xb.x, xb.y, xb.z, xb.w};

            #pragma unroll
            for (int c = 0; c < CO_; ++c) {
                // W(kw,c) = component (c&3) of wq[2*kw + (c>>2)] -- constant after unroll
                #define WTAP(kw, c) (((const float*)&wq[2*(kw) + ((c) >> 2)])[(c) & 3])
                #pragma unroll
                for (int j = 0; j < 4; ++j) {
                    float a0 = xr[j + 0] + WTAP(0, c);
                    float a1 = xr[j + 1] + WTAP(1, c);
                    float a2 = xr[j + 2] + WTAP(2, c);
                    float a3 = xr[j + 3] + WTAP(3, c);
                    float a4 = xr[j + 4] + WTAP(4, c);
                    // shaped for v_max3_num_f32 folding:
                    float m012 = fmaxf(fmaxf(a0, a1), a2);            // -> v_max3
                    float m34  = fmaxf(a3, a4);                       // -> v_max
                    acc[c][j]  = fmaxf(fmaxf(acc[c][j], m012), m34);  // -> v_max3
                }
                #undef WTAP
            }
        }
    }

    // ---- store: one float4 per co per thread, coalesced across lanes
    const int h = h0 + ty;
    #pragma unroll
    for (int c = 0; c < CO_; ++c) {
        float4 v = make_float4(acc[c][0], acc[c][1], acc[c][2], acc[c][3]);
        *(float4*)(&out[(((size_t)n * CO_ + c) * H_ + h) * W_ + (w0 + 4 * txq)]) = v;
    }
}

extern "C" void kernel_launch(void* const* d_in, const int* in_sizes, int n_in,
                              void* d_out, int out_size, void* d_ws, size_t ws_size,
                              hipStream_t stream) {
    const float* x  = (const float*)d_in[0];   // (8,4,512,512) f32
    const float* wt = (const float*)d_in[1];   // (8,4,5,5)     f32
    float* out      = (float*)d_out;           // (8,8,512,512) f32

    dim3 grid(W_ / TW_, H_ / TH_, N_);         // 8 x 32 x 8 = 2048 blocks
    dilation2d_maxplus_kernel<<<grid, NTHREADS, 0, stream>>>(x, wt, out);
}